// PolyGNN_18872086299269
// MI455X (gfx1250) — compile-verified
//
#include <hip/hip_runtime.h>

// PolyGNN on gfx1250 (MI455X), fp32 WMMA (V_WMMA_F32_16X16X4_F32).
// Activations stored padded [8192 x 128] f32; padding columns kept zero.
// adj@x exploits the banded cyclic adjacency (n_adj=20 -> band +-10 < 16):
// for a 16-row tile only the 48-col window [m0-16, m0+32) mod 512 can be
// non-zero, and we read the *actual* adj values in that window (exact for
// any adjacency of bandwidth <= 16, which the reference guarantees).
// GEMM shapes are compile-time (templates) so K-loops fully unroll into
// straight-line v_wmma chains with pipelined global loads.

typedef float v2f __attribute__((ext_vector_type(2)));
typedef float v8f __attribute__((ext_vector_type(8)));

#define BATCH 16
#define NNODE 512
#define CCH   120
#define HH    96
#define WW    96
#define HWSZ  (HH * WW)
#define FPAD  128
#define ROWS  (BATCH * NNODE)   // 8192
#define NRESB 6
#define FDIM  122
#define HID   120

// ---------------------------------------------------------------- gather (s=0)
__global__ __launch_bounds__(256) void k_gather0(
    const float* __restrict__ tg2, const float* __restrict__ hull,
    const int* __restrict__ idx, float* __restrict__ X) {
  int t = blockIdx.x * blockDim.x + threadIdx.x;           // ROWS*FPAD threads
  if (t >= ROWS * FPAD) return;
  int c = t & (FPAD - 1);
  int row = t >> 7;
  float v = 0.0f;
  if (c < CCH) {
    int b = row >> 9;                                      // row / 512
    int p = idx[row];
    v = tg2[(size_t)b * CCH * HWSZ + (size_t)c * HWSZ + p];
  } else if (c < CCH + 2) {
    v = hull[row * 2 + (c - CCH)];
  }
  X[t] = v;
}

// ------------------------------------------------------- bilinear gather (s=1)
__global__ __launch_bounds__(256) void k_gather_interp(
    const float* __restrict__ tg2, const float* __restrict__ poly,
    float* __restrict__ X) {
  int t = blockIdx.x * blockDim.x + threadIdx.x;
  if (t >= ROWS * FPAD) return;
  int c = t & (FPAD - 1);
  int row = t >> 7;
  float v = 0.0f;
  if (c < CCH) {
    int b = row >> 9;
    float cx = poly[row * 2 + 0] * (float)HH;
    float cy = poly[row * 2 + 1] * (float)WW;
    float X0 = floorf(cx), Y0 = floorf(cy);
    float X1 = X0 + 1.0f, Y1 = Y0 + 1.0f;
    // weights computed pre-clamp (faithful to reference)
    float w00 = (X1 - cx) * (Y1 - cy);
    float w01 = (X1 - cx) * (cy - Y0);
    float w10 = (cx - X0) * (Y1 - cy);
    float w11 = (cx - X0) * (cy - Y0);
    int x0 = (int)fminf(fmaxf(X0, 0.0f), (float)(HH - 1));
    int x1 = (int)fminf(fmaxf(X1, 0.0f), (float)(HH - 1));
    int y0 = (int)fminf(fmaxf(Y0, 0.0f), (float)(WW - 1));
    int y1 = (int)fminf(fmaxf(Y1, 0.0f), (float)(WW - 1));
    const float* fb = tg2 + (size_t)b * CCH * HWSZ + (size_t)c * HWSZ;
    v = w00 * fb[x0 * WW + y0] + w01 * fb[x0 * WW + y1] +
        w10 * fb[x1 * WW + y0] + w11 * fb[x1 * WW + y1];
  } else if (c < CCH + 2) {
    v = poly[row * 2 + (c - CCH)];
  }
  X[t] = v;
}

// -------------------------------------------------- banded adj @ X via WMMA f32
// one wave per 16x16 output tile; tiles = BATCH * 32 * 8 = 4096
__global__ __launch_bounds__(128) void k_adjmm(
    const float* __restrict__ adj, const float* __restrict__ X,
    float* __restrict__ AX) {
  int tile = blockIdx.x * (blockDim.x >> 5) + (threadIdx.x >> 5);
  int nt = tile & 7;
  int mt = (tile >> 3) & 31;
  int b  = tile >> 8;
  int lane = threadIdx.x & 31;
  int half = lane >> 4;        // 0: K 0..1 / M 0..7 ; 1: K 2..3 / M 8..15
  int l16  = lane & 15;
  int koff = half * 2;
  int m0 = mt * 16, n0 = nt * 16;
  const float* adjb = adj + (size_t)b * NNODE * NNODE;
  const float* Xb   = X   + (size_t)b * NNODE * FPAD;
  v8f acc = {};
#pragma unroll
  for (int blk = 0; blk < 3; ++blk) {
    int kb = (m0 + (blk - 1) * 16 + NNODE) & (NNODE - 1);
#pragma unroll
    for (int k = 0; k < 16; k += 4) {
      int kk = kb + k + koff;
      v2f a, bv;
      a.x = adjb[(m0 + l16) * NNODE + kk];
      a.y = adjb[(m0 + l16) * NNODE + kk + 1];
      bv.x = Xb[(size_t)kk * FPAD + n0 + l16];
      bv.y = Xb[(size_t)(kk + 1) * FPAD + n0 + l16];
      acc = __builtin_amdgcn_wmma_f32_16x16x4_f32(
          false, a, false, bv, (short)0, acc, false, false);
    }
  }
  float* out = AX + (size_t)b * NNODE * FPAD;
#pragma unroll
  for (int v = 0; v < 8; ++v)
    out[(size_t)(m0 + half * 8 + v) * FPAD + n0 + l16] = acc[v];
}

// ------------------------- fused gconv: OUT = act(X@W1 + AX@W2 + b1+b2 [+res])
// MODE: 0 = linear, 1 = relu, 2 = relu(out + resid)
// KV/NV compile-time -> full unroll, guards fold away.
template <int KV, int NV, int MODE>
__global__ __launch_bounds__(128) void k_gconv_t(
    const float* __restrict__ X, const float* __restrict__ AX,
    const float* __restrict__ W1, const float* __restrict__ W2,
    const float* __restrict__ b1, const float* __restrict__ b2,
    const float* __restrict__ resid, float* __restrict__ OUT) {
  int tile = blockIdx.x * (blockDim.x >> 5) + (threadIdx.x >> 5);
  int nt = tile & 7;
  int mt = tile >> 3;                       // 0..511
  int lane = threadIdx.x & 31;
  int half = lane >> 4;
  int l16  = lane & 15;
  int koff = half * 2;
  int m0 = mt * 16, n0 = nt * 16;
  int nn = n0 + l16;
  bool okn = nn < NV;
  constexpr int KPAD = (KV + 3) & ~3;
  v8f acc = {};
#pragma unroll
  for (int k = 0; k < KPAD; k += 4) {
    int kk = k + koff;
    v2f a, bv;
    a.x = X[(size_t)(m0 + l16) * FPAD + kk];
    a.y = X[(size_t)(m0 + l16) * FPAD + kk + 1];
    bv.x = (okn && kk < KV)     ? W1[(size_t)kk * NV + nn]       : 0.0f;
    bv.y = (okn && kk + 1 < KV) ? W1[(size_t)(kk + 1) * NV + nn] : 0.0f;
    acc = __builtin_amdgcn_wmma_f32_16x16x4_f32(
        false, a, false, bv, (short)0, acc, false, false);
  }
#pragma unroll
  for (int k = 0; k < KPAD; k += 4) {
    int kk = k + koff;
    v2f a, bv;
    a.x = AX[(size_t)(m0 + l16) * FPAD + kk];
    a.y = AX[(size_t)(m0 + l16) * FPAD + kk + 1];
    bv.x = (okn && kk < KV)     ? W2[(size_t)kk * NV + nn]       : 0.0f;
    bv.y = (okn && kk + 1 < KV) ? W2[(size_t)(kk + 1) * NV + nn] : 0.0f;
    acc = __builtin_amdgcn_wmma_f32_16x16x4_f32(
        false, a, false, bv, (short)0, acc, false, false);
  }
  float bias = okn ? (b1[nn] + b2[nn]) : 0.0f;
#pragma unroll
  for (int v = 0; v < 8; ++v) {
    int r = m0 + half * 8 + v;
    float o = acc[v] + bias;
    if (MODE == 2) o += resid[(size_t)r * FPAD + n0 + l16];
    if (MODE >= 1) o = fmaxf(o, 0.0f);
    OUT[(size_t)r * FPAD + n0 + l16] = o;
  }
}

// ----------------------------------------- fc (32->2) + poly update + output
__global__ __launch_bounds__(256) void k_fc_poly(
    const float* __restrict__ Y, const float* __restrict__ fcW,
    const float* __restrict__ fcb, const float* __restrict__ polyPrev,
    float* __restrict__ polyOut) {
  int row = blockIdx.x * blockDim.x + threadIdx.x;
  if (row >= ROWS) return;
  float s0 = fcb[0], s1 = fcb[1];
#pragma unroll
  for (int j = 0; j < 32; ++j) {
    float y = Y[(size_t)row * FPAD + j];
    s0 += y * fcW[j * 2 + 0];
    s1 += y * fcW[j * 2 + 1];
  }
  polyOut[row * 2 + 0] = polyPrev[row * 2 + 0] + s0;
  polyOut[row * 2 + 1] = polyPrev[row * 2 + 1] + s1;
}

// ---------------------------------------------------------------------- launch
extern "C" void kernel_launch(void* const* d_in, const int* in_sizes, int n_in,
                              void* d_out, int out_size, void* d_ws, size_t ws_size,
                              hipStream_t stream) {
  const float* tg2  = (const float*)d_in[0];
  const float* hull = (const float*)d_in[1];
  const float* adj  = (const float*)d_in[2];
  const float* g0W1 = (const float*)d_in[3];
  const float* g0b1 = (const float*)d_in[4];
  const float* g0W2 = (const float*)d_in[5];
  const float* g0b2 = (const float*)d_in[6];
  const float* resW = (const float*)d_in[7];
  const float* resb = (const float*)d_in[8];
  const float* g7W1 = (const float*)d_in[9];
  const float* g7b1 = (const float*)d_in[10];
  const float* g7W2 = (const float*)d_in[11];
  const float* g7b2 = (const float*)d_in[12];
  const float* fcW  = (const float*)d_in[13];
  const float* fcb  = (const float*)d_in[14];
  const int*   pidx = (const int*)d_in[15];
  float* out = (float*)d_out;

  const size_t BUF = (size_t)ROWS * FPAD;     // 1M floats
  float* P0 = (float*)d_ws;                    // activations / temps
  float* P1 = P0 + BUF;                        // adj@x
  float* P2 = P1 + BUF;                        // activations

  const int elemGrid = (ROWS * FPAD + 255) / 256;  // 4096
  const int tileGrid = 4096 / 4;                   // 1024 blocks x 4 waves

  for (int s = 0; s < 2; ++s) {
    float* poly_prev = (s == 0) ? (float*)hull : out;       // [8192,2]
    float* poly_out  = out + (size_t)s * ROWS * 2;

    // build input features -> P0
    if (s == 0)
      k_gather0<<<elemGrid, 256, 0, stream>>>(tg2, hull, pidx, P0);
    else
      k_gather_interp<<<elemGrid, 256, 0, stream>>>(tg2, out /*poly0*/, P0);

    // gcn0: 122 -> 120, linear
    k_adjmm<<<tileGrid, 128, 0, stream>>>(adj, P0, P1);
    k_gconv_t<FDIM, HID, 0><<<tileGrid, 128, 0, stream>>>(
        P0, P1,
        g0W1 + (size_t)s * FDIM * HID, g0W2 + (size_t)s * FDIM * HID,
        g0b1 + (size_t)s * HID,        g0b2 + (size_t)s * HID,
        nullptr, P2);
    float* cur = P2;   // 120-wide activations

    // 6 residual blocks
    for (int r = 0; r < NRESB; ++r) {
      const float* W00 = resW + (size_t)((((s * NRESB) + r) * 2 + 0) * 2 + 0) * HID * HID;
      const float* W01 = resW + (size_t)((((s * NRESB) + r) * 2 + 0) * 2 + 1) * HID * HID;
      const float* W10 = resW + (size_t)((((s * NRESB) + r) * 2 + 1) * 2 + 0) * HID * HID;
      const float* W11 = resW + (size_t)((((s * NRESB) + r) * 2 + 1) * 2 + 1) * HID * HID;
      const float* b00 = resb + (size_t)((((s * NRESB) + r) * 2 + 0) * 2 + 0) * HID;
      const float* b01 = resb + (size_t)((((s * NRESB) + r) * 2 + 0) * 2 + 1) * HID;
      const float* b10 = resb + (size_t)((((s * NRESB) + r) * 2 + 1) * 2 + 0) * HID;
      const float* b11 = resb + (size_t)((((s * NRESB) + r) * 2 + 1) * 2 + 1) * HID;

      // h = relu(gconv(x))
      k_adjmm<<<tileGrid, 128, 0, stream>>>(adj, cur, P1);
      k_gconv_t<HID, HID, 1><<<tileGrid, 128, 0, stream>>>(
          cur, P1, W00, W01, b00, b01, nullptr, P0);
      // x = relu(gconv(h) + x)
      k_adjmm<<<tileGrid, 128, 0, stream>>>(adj, P0, P1);
      k_gconv_t<HID, HID, 2><<<tileGrid, 128, 0, stream>>>(
          P0, P1, W10, W11, b10, b11, cur, cur);
    }

    // gcn7: 120 -> 32, linear -> P0
    k_adjmm<<<tileGrid, 128, 0, stream>>>(adj, cur, P1);
    k_gconv_t<HID, 32, 0><<<tileGrid, 128, 0, stream>>>(
        cur, P1,
        g7W1 + (size_t)s * HID * 32, g7W2 + (size_t)s * HID * 32,
        g7b1 + (size_t)s * 32,       g7b2 + (size_t)s * 32,
        nullptr, P0);

    // fc 32->2, poly_out = poly_prev + shift
    k_fc_poly<<<(ROWS + 255) / 256, 256, 0, stream>>>(
        P0, fcW + (size_t)s * 64, fcb + (size_t)s * 2, poly_prev, poly_out);
  }
}